// CTCLoss_81561428951535
// MI455X (gfx1250) — compile-verified
//
#include <hip/hip_runtime.h>
#include <math.h>

typedef __attribute__((ext_vector_type(2))) float v2f;
typedef __attribute__((ext_vector_type(8))) float v8f;

#define NEGV (-1e30f)

// Precise logaddexp (used only in the cheap finalize step)
__device__ __forceinline__ float lae_precise(float x, float y) {
    float m = fmaxf(x, y);
    float d = fminf(x, y) - m;          // <= 0
    return m + log1pf(expf(d));
}

// Fast logaddexp for the latency-bound scan: one v_exp_f32 + one v_log_f32.
// z = exp(d) in (0,1]; log(1+z) vs log1p(z) differ only for z < ~1e-7 where
// the absolute contribution is itself < 1e-7 in the log domain -> negligible.
__device__ __forceinline__ float lae(float x, float y) {
    float m = fmaxf(x, y);
    float d = fminf(x, y) - m;          // <= 0 (can be ~-1e30 -> exp -> 0)
    return m + __logf(1.0f + __expf(d));
}

// ---------------------------------------------------------------------------
// Kernel 1: per-(n,t) row max + log-sum-exp over C=1024.
// 1 block / row, 256 threads, each thread owns 4 contiguous floats (b128 load).
// Bandwidth-bound: 256 MB streamed once (~11us floor at 23.3 TB/s); precise
// expf VALU work overlaps the stream. Sum-of-exp wave reduction runs on the
// matrix pipe via V_WMMA_F32_16X16X4_F32 with an all-ones B matrix (exact f32).
// ---------------------------------------------------------------------------
__global__ __launch_bounds__(256) void rowstats_kernel(
    const float* __restrict__ pred, float* __restrict__ rowstats, int C) {
    const int row = blockIdx.x;               // n*T + t
    const int tid = threadIdx.x;
    const float4* rp = (const float4*)(pred + (size_t)row * (size_t)C);
    float4 v = rp[tid];

    // --- block max ---
    float m = fmaxf(fmaxf(v.x, v.y), fmaxf(v.z, v.w));
    #pragma unroll
    for (int off = 16; off > 0; off >>= 1)
        m = fmaxf(m, __shfl_xor(m, off, 32));

    __shared__ float s_max[8];
    __shared__ float s_sum[8];
    const int wave = tid >> 5, lane = tid & 31;
    if (lane == 0) s_max[wave] = m;
    __syncthreads();
    float bmax = s_max[0];
    #pragma unroll
    for (int w = 1; w < 8; ++w) bmax = fmaxf(bmax, s_max[w]);

    // --- per-thread partial sum of exp (precise; overlapped with HBM stream) ---
    float p = expf(v.x - bmax) + expf(v.y - bmax) +
              expf(v.z - bmax) + expf(v.w - bmax);

    // --- intra-wave sum via WMMA: D = A(16x4) * ones(4x16) ---
    // A: lane l<16 holds {A[l,0]=p, A[l,1]=0}; lane l>=16 holds {A[l-16,2]=p, 0}
    // => D[m,n] = p_m + p_{m+16}. Sum 8 D regs + shfl_xor(16) = full 32-lane sum.
    v2f a; a.x = p;   a.y = 0.0f;
    v2f b; b.x = 1.0f; b.y = 1.0f;
    v8f c = {};
    c = __builtin_amdgcn_wmma_f32_16x16x4_f32(false, a, false, b, (short)0, c,
                                              false, false);
    float wsum = c[0] + c[1] + c[2] + c[3] + c[4] + c[5] + c[6] + c[7];
    wsum += __shfl_xor(wsum, 16, 32);

    if (lane == 0) s_sum[wave] = wsum;
    __syncthreads();
    if (tid == 0) {
        float tot = 0.0f;
        #pragma unroll
        for (int w = 0; w < 8; ++w) tot += s_sum[w];
        rowstats[2 * row]     = bmax;
        rowstats[2 * row + 1] = logf(tot);
    }
}

// ---------------------------------------------------------------------------
// Kernel 2: CTC alpha recursion. 1 block per batch element, thread s owns
// extended state s (S2 = 2S+1 = 257). Alpha double-buffered in LDS -> single
// barrier per time step. Latency-bound serial chain of T steps, so:
//  - t+1 gathered logit + row stats prefetched into registers (1 ahead)
//  - t+2 gather line prefetched into cache via global_prefetch (2 ahead)
//  - logaddexp uses hardware v_exp_f32 / v_log_f32 (short dependency chain)
// ---------------------------------------------------------------------------
__global__ __launch_bounds__(512) void alpha_kernel(
    const float* __restrict__ pred, const int* __restrict__ gt,
    const int* __restrict__ plen, const int* __restrict__ glen,
    const float* __restrict__ rowstats, float* __restrict__ losses,
    int T, int C, int S) {
    const int n  = blockIdx.x;
    const int S2 = 2 * S + 1;
    const int s  = threadIdx.x;
    const bool active = (s < S2);
    const int sc = active ? s : 0;

    // extended label: blank(0) at even s, gt[(s-1)/2] at odd s
    int ext_s = (sc & 1) ? gt[n * S + (sc >> 1)] : 0;
    // skip transition allowed iff ext[s]!=blank && ext[s]!=ext[s-2]
    bool skip = false;
    if ((sc & 1) && ext_s != 0) {
        if (sc == 1) skip = true;                           // ext_m2 == -1
        else skip = (ext_s != gt[n * S + ((sc - 2) >> 1)]);
    }
    const int myplen = plen[n];

    __shared__ float sA[512];
    __shared__ float sB[512];

    const float*  gptr  = pred + (size_t)n * (size_t)T * (size_t)C + ext_s;
    const float2* rptr2 = (const float2*)(rowstats + (size_t)n * (size_t)T * 2);

    // t = 0 init: only s=0,1 reachable
    float2 rs0 = rptr2[0];
    float lp0 = gptr[0] - rs0.x - rs0.y;
    float aSelf = (active && sc < 2) ? lp0 : NEGV;
    if (active) sA[sc] = aSelf;

    // prefetch t = 1
    float  nextRaw = gptr[(size_t)C];
    float2 nextRS  = rptr2[1];
    __syncthreads();

    float* rd = sA;
    float* wr = sB;
    for (int t = 1; t < T; ++t) {
        float  curRaw = nextRaw;
        float2 curRS  = nextRS;
        if (t + 1 < T) {                                    // register prefetch t+1
            nextRaw = gptr[(size_t)(t + 1) * (size_t)C];
            nextRS  = rptr2[t + 1];
        }
        if (t + 2 < T)                                      // cache prefetch t+2
            __builtin_prefetch(&gptr[(size_t)(t + 2) * (size_t)C], 0, 0);

        float lp = curRaw - curRS.x - curRS.y;
        float a1 = (sc >= 1) ? rd[sc - 1] : NEGV;
        float a2 = (skip && sc >= 2) ? rd[sc - 2] : NEGV;
        float nv = lae(lae(aSelf, a1), a2) + lp;
        float aNew = (t < myplen) ? nv : aSelf;             // freeze past length
        if (active) wr[sc] = aNew;
        aSelf = aNew;
        __syncthreads();                                    // single barrier (dbl buffer)
        float* tmp = rd; rd = wr; wr = tmp;
    }

    if (s == 0) {
        int L  = glen[n];
        int ib = 2 * L;
        int il = (ib > 0) ? ib - 1 : 0;
        float ll = lae_precise(rd[ib], rd[il]);
        float loss = -ll;
        if (loss > 1e29f) loss = 0.0f;                      // zero_infinity
        float denom = (float)(L > 0 ? L : 1);
        losses[n] = loss / denom;
    }
}

// ---------------------------------------------------------------------------
// Kernel 3: batch mean
// ---------------------------------------------------------------------------
__global__ void mean_kernel(const float* __restrict__ losses,
                            float* __restrict__ out, int N) {
    if (threadIdx.x == 0) {
        float acc = 0.0f;
        for (int i = 0; i < N; ++i) acc += losses[i];
        out[0] = acc / (float)N;
    }
}

extern "C" void kernel_launch(void* const* d_in, const int* in_sizes, int n_in,
                              void* d_out, int out_size, void* d_ws, size_t ws_size,
                              hipStream_t stream) {
    const float* pred = (const float*)d_in[0];
    const int*   gt   = (const int*)d_in[1];
    const int*   plen = (const int*)d_in[2];
    const int*   glen = (const int*)d_in[3];

    const int N = in_sizes[2];                       // 64
    const int S = in_sizes[1] / N;                   // 128
    const int C = 1024;                              // fixed per reference
    const int T = (int)((long long)in_sizes[0] / ((long long)N * C));  // 1024

    float* rowstats = (float*)d_ws;                  // N*T*2 f32 = 512 KB
    float* losses   = rowstats + (size_t)N * T * 2;  // N f32

    rowstats_kernel<<<N * T, 256, 0, stream>>>(pred, rowstats, C);

    const int S2 = 2 * S + 1;
    int bdim = ((S2 + 31) / 32) * 32;                // 288 threads (9 waves)
    if (bdim > 512) bdim = 512;
    alpha_kernel<<<N, bdim, 0, stream>>>(pred, gt, plen, glen, rowstats, losses,
                                         T, C, S);

    mean_kernel<<<1, 32, 0, stream>>>(losses, (float*)d_out, N);
}